// EncodecWrapper_23759759081966
// MI455X (gfx1250) — compile-verified
//
#include <hip/hip_runtime.h>

typedef float v2f __attribute__((ext_vector_type(2)));
typedef float v8f __attribute__((ext_vector_type(8)));

#define B_ 16
#define D_ 128
#define T_ 4096
#define V_ 1024

__global__ __launch_bounds__(256) void vq_argmin_kernel(
    const float* __restrict__ latents,   // (B, D, T)
    const float* __restrict__ codebook,  // (V, D)
    int* __restrict__ codes_out,         // (B, T)
    float* __restrict__ q_out)           // (B, D, T)
{
    __shared__ float As[D_][16];      // x tile, [k][m] : lanes read 16 consecutive m -> no bank conflict
    __shared__ float candVal[8][16];  // per-wave best value per row m
    __shared__ int   candIdx[8][16];
    __shared__ int   codesS[16];

    const int tid  = threadIdx.x;
    const int lane = tid & 31;
    const int wv   = tid >> 5;        // wave32 id, 8 waves
    const int n    = lane & 15;       // column (N) this lane owns in B/C/D
    const bool hi  = lane >= 16;
    const int khi  = hi ? 2 : 0;      // per-lane K residue base (B layout + A layout)

    const int tileId = blockIdx.x;        // B * T/16 = 4096 tiles
    const int b  = tileId >> 8;           // 256 t-tiles per batch
    const int t0 = (tileId & 255) << 4;

    // Stage x tile: As[d][m] = latents[b, d, t0+m]  (coalesced along t)
    {
        const float* src = latents + (size_t)b * D_ * T_ + t0;
        const int m = tid & 15;
        const int dbase = tid >> 4;
#pragma unroll
        for (int i = 0; i < 8; ++i) {
            const int d = dbase + i * 16;
            As[d][m] = src[(size_t)d * T_ + m];
        }
    }
    __syncthreads();

    float bestVal[8];
    int   bestIdx[8];
#pragma unroll
    for (int r = 0; r < 8; ++r) { bestVal[r] = 3.4e38f; bestIdx[r] = 0x7fffffff; }

    const int m = n;  // A-row this lane feeds (ISA: A 16x4, M = lane & 15)

    // Each wave handles 8 contiguous V-tiles of 16 codes
    for (int j = 0; j < 8; ++j) {
        const int v0 = (wv * 8 + j) * 16;
        // Per-lane pair base: lanes 0-15 need K = 4kc+{0,1}, lanes 16-31 need K = 4kc+{2,3}.
        // Fold the +2 into the pointer once -> plain b64 loads, no per-element selects.
        const float* cb = codebook + (size_t)(v0 + n) * D_ + khi;

        v8f acc = {0.f, 0.f, 0.f, 0.f, 0.f, 0.f, 0.f, 0.f};
        float csq_part = 0.f;
#pragma unroll 8
        for (int kc = 0; kc < 32; ++kc) {
            const float2 f = *(const float2*)(cb + kc * 4);   // global_load_b64
            csq_part += f.x * f.x + f.y * f.y;                // half of |c|^2 per half-wave
            v2f bvec; bvec.x = f.x; bvec.y = f.y;
            v2f avec;
            const int kb = kc * 4 + khi;
            avec.x = As[kb + 0][m];
            avec.y = As[kb + 1][m];
            acc = __builtin_amdgcn_wmma_f32_16x16x4_f32(
                false, avec, false, bvec, (short)0, acc, false, false);
        }
        // complete |c|^2: partner half-wave has the other K residues
        const float csq = csq_part + __shfl_xor(csq_part, 16, 32);
        const int code = v0 + n;
#pragma unroll
        for (int r = 0; r < 8; ++r) {
            const float s = csq - 2.0f * acc[r];    // argmin-equivalent distance
            if (s < bestVal[r]) { bestVal[r] = s; bestIdx[r] = code; }
        }
    }

    // Reduce across the 16 N-lanes (tie -> lower code index, matching argmin)
#pragma unroll
    for (int off = 8; off >= 1; off >>= 1) {
#pragma unroll
        for (int r = 0; r < 8; ++r) {
            const float ov = __shfl_xor(bestVal[r], off, 32);
            const int   oi = __shfl_xor(bestIdx[r], off, 32);
            if (ov < bestVal[r] || (ov == bestVal[r] && oi < bestIdx[r])) {
                bestVal[r] = ov; bestIdx[r] = oi;
            }
        }
    }
    // lane 0 holds rows m=0..7, lane 16 holds m=8..15 (C/D layout)
    if (n == 0) {
        const int mbase = hi ? 8 : 0;
#pragma unroll
        for (int r = 0; r < 8; ++r) {
            candVal[wv][mbase + r] = bestVal[r];
            candIdx[wv][mbase + r] = bestIdx[r];
        }
    }
    __syncthreads();

    // Cross-wave reduction + emit codes
    if (tid < 16) {
        float bv = candVal[0][tid];
        int   bi = candIdx[0][tid];
#pragma unroll
        for (int w = 1; w < 8; ++w) {
            const float ov = candVal[w][tid];
            const int   oi = candIdx[w][tid];
            if (ov < bv || (ov == bv && oi < bi)) { bv = ov; bi = oi; }
        }
        codesS[tid] = bi;
        codes_out[(size_t)b * T_ + t0 + tid] = bi;
    }
    __syncthreads();

    // quantized[b, d, t0+tl] = codebook[code[tl], d]  (coalesced along t)
    {
        const int tl = tid & 15;
        const int dbase = tid >> 4;
        const float* crow = codebook + (size_t)codesS[tl] * D_;
        float* qb = q_out + (size_t)b * D_ * T_ + t0 + tl;
#pragma unroll
        for (int i = 0; i < 8; ++i) {
            const int d = i * 16 + dbase;
            qb[(size_t)d * T_] = crow[d];
        }
    }
}

extern "C" void kernel_launch(void* const* d_in, const int* in_sizes, int n_in,
                              void* d_out, int out_size, void* d_ws, size_t ws_size,
                              hipStream_t stream) {
    (void)in_sizes; (void)n_in; (void)out_size; (void)d_ws; (void)ws_size;
    const float* latents  = (const float*)d_in[0];   // B*D*T f32
    const float* codebook = (const float*)d_in[1];   // V*D   f32
    int*   codes_out = (int*)d_out;                          // first B*T int32
    float* q_out     = (float*)d_out + (size_t)B_ * T_;      // then B*D*T f32

    const int grid = B_ * (T_ / 16);   // 4096 blocks, one 16-row t-tile each
    vq_argmin_kernel<<<grid, 256, 0, stream>>>(latents, codebook, codes_out, q_out);
}